// SpatiallyVaryingCrossChannelConvolution_23424751633063
// MI455X (gfx1250) — compile-verified
//
#include <hip/hip_runtime.h>
#include <stdint.h>

// ---------------------------------------------------------------------------
// Spatially-varying cross-channel 7x7 convolution, reflect padding.
//   out[o,y,x] = (1/3) * sum_c sum_{dy,dx} x_ref[c, y+dy-3, x+dx-3] * K[o,c,y,x,dy,dx]
// K is 462 MB and consumed exactly once -> pure HBM streaming problem
// (~20 us floor at 23.3 TB/s). We stream K with the CDNA5 Tensor Data Mover
// into LDS (double buffered, TENSORcnt-tracked), and keep the tiny reusable
// x window in LDS with reflect handled once at staging time.
// ---------------------------------------------------------------------------

typedef __attribute__((ext_vector_type(4))) uint32_t v4u;
typedef __attribute__((ext_vector_type(8))) int      v8i;
typedef __attribute__((ext_vector_type(4))) int      v4i;

constexpr int IMG_H   = 512;
constexpr int IMG_W   = 512;
constexpr int INC     = 3;
constexpr int OUTC    = 3;
constexpr int KH      = 7;
constexpr int KW      = 7;
constexpr int TAPS    = KH * KW;            // 49
constexpr int BX      = 64;                 // pixels per block (x)
constexpr int TILE    = BX * TAPS;          // 3136 floats: one o-row of a c-stage
constexpr int KSTAGE  = OUTC * TILE;        // 9408 floats per c-stage
constexpr int XSW     = BX + KW - 1;        // 70 staged x columns
constexpr int XSN     = INC * KH * XSW;     // 1470 staged x floats
constexpr int NTHREADS = OUTC * BX;         // 192 threads = 6 waves

// o-dimension stride inside K (elements): ((o+1)*INC) plane vs (o*INC) plane
constexpr int O_STRIDE_ELEMS = INC * IMG_H * IMG_W * TAPS;   // 38,535,168

// Issue one TDM 2-D tile load: 3136 contiguous f32 (dim0) x 3 rows (dim1, the
// o-channels) with 48-bit row stride, into LDS at lds_byte.
__device__ __forceinline__ void tdm_load_ktile(const float* gsrc, uint32_t lds_byte) {
    const uint64_t ga = (uint64_t)(uintptr_t)gsrc;

    // D# group 0 (128b): count=1 valid user descriptor, lds_addr, global_addr, type=2
    v4u g0;
    g0.x = 1u;                                            // [1:0] count = 1
    g0.y = lds_byte;                                      // [63:32] lds_addr (bytes)
    g0.z = (uint32_t)(ga & 0xFFFFFFFFull);                // global_addr[31:0]
    g0.w = (uint32_t)((ga >> 32) & 0x1FFFFFFull)          // global_addr[56:32]
         | (2u << 30);                                    // [127:126] type = 2 (image)

    // D# group 1 (256b)
    v8i g1;
    g1[0] = (int)(2u << 16);                              // data_size=2 (4B); wg_mask=0
    g1[1] = (int)((uint32_t)TILE << 16);                  // tensor_dim0[15:0]=3136
    g1[2] = (int)((uint32_t)OUTC << 16);                  // tensor_dim0 hi=0 | tensor_dim1 lo=3
    g1[3] = (int)((uint32_t)TILE << 16);                  // tensor_dim1 hi=0 | tile_dim0=3136
    g1[4] = OUTC;                                         // tile_dim1=3, tile_dim2=0
    g1[5] = O_STRIDE_ELEMS;                               // tensor_dim0_stride[31:0]
    g1[6] = 0;                                            // stride0 hi=0 | stride1 lo=0
    g1[7] = 0;                                            // stride1 hi=0

    const v4i z4 = {0, 0, 0, 0};                          // groups 2/3 unused (<=2D)
    const v8i z8 = {0, 0, 0, 0, 0, 0, 0, 0};              // extra words (clang-23 form)
    __builtin_amdgcn_tensor_load_to_lds(g0, g1, z4, z4, z8, 0); // cpol = default
}

__global__ __launch_bounds__(NTHREADS)
void svconv_kernel(const float* __restrict__ x,
                   const float* __restrict__ kern,
                   float* __restrict__ out) {
    __shared__ float lds[2 * KSTAGE + XSN];               // 81,144 bytes

    const int y   = blockIdx.y;                           // 0..511
    const int x0  = blockIdx.x * BX;                      // 0..448, multiple of 64
    const int tid = (int)threadIdx.x;
    // wave-uniform wave id -> scalar branch so exactly one wave drives the TDM
    const int wid = __builtin_amdgcn_readfirstlane(tid) >> 5;

    // element offset of this block's contiguous K row-segment for (o=0, c)
    const size_t base    = ((size_t)y * IMG_W + (size_t)x0) * TAPS;
    const size_t cstride = (size_t)IMG_H * IMG_W * TAPS;  // c-plane stride (elements)

    // Kick off stage c=0 DMA immediately (overlaps with x staging below).
    if (wid == 0) {
        tdm_load_ktile(kern + base, (uint32_t)(uintptr_t)&lds[0]);
    }

    // Stage the x window (reflect padding resolved here) into LDS.
    for (int i = tid; i < XSN; i += NTHREADS) {
        const int c = i / (KH * XSW);
        const int r = (i / XSW) % KH;
        const int j = i % XSW;
        int ry = y - KH / 2 + r;
        ry = ry < 0 ? -ry : (ry > IMG_H - 1 ? 2 * (IMG_H - 1) - ry : ry);
        int rx = x0 - KW / 2 + j;
        rx = rx < 0 ? -rx : (rx > IMG_W - 1 ? 2 * (IMG_W - 1) - rx : rx);
        lds[2 * KSTAGE + i] = x[((size_t)c * IMG_H + ry) * IMG_W + rx];
    }

    const int o  = tid / BX;                              // 0..2
    const int px = tid % BX;                              // 0..63
    float acc = 0.0f;

#pragma unroll
    for (int c = 0; c < INC; ++c) {
        if (wid == 0) {
            if (c + 1 < INC) {
                // prefetch next c-stage into the other buffer
                tdm_load_ktile(kern + base + (size_t)(c + 1) * cstride,
                               (uint32_t)(uintptr_t)&lds[((c + 1) & 1) * KSTAGE]);
                // TDM ops complete in order: <=1 outstanding => stage c arrived
                __builtin_amdgcn_s_wait_tensorcnt(1);
            } else {
                __builtin_amdgcn_s_wait_tensorcnt(0);
            }
        }
        __syncthreads();  // K stage c (and, on c==0, the x window) now visible

        const float* kb = &lds[(c & 1) * KSTAGE + o * TILE + px * TAPS];
        const float* xc = &lds[2 * KSTAGE + c * (KH * XSW)];
#pragma unroll
        for (int dy = 0; dy < KH; ++dy) {
#pragma unroll
            for (int dx = 0; dx < KW; ++dx) {
                acc = __builtin_fmaf(xc[dy * XSW + px + dx], kb[dy * KW + dx], acc);
            }
        }
        __syncthreads();  // all waves done reading buffer (c&1) before its reuse
    }

    out[((size_t)o * IMG_H + y) * IMG_W + x0 + px] = acc * (1.0f / 3.0f);
}

extern "C" void kernel_launch(void* const* d_in, const int* in_sizes, int n_in,
                              void* d_out, int out_size, void* d_ws, size_t ws_size,
                              hipStream_t stream) {
    const float* x    = (const float*)d_in[0];   // [1,3,512,512]
    const float* kern = (const float*)d_in[1];   // [1,3,3,512,512,7,7]
    float* out        = (float*)d_out;           // [1,3,512,512]

    dim3 grid(IMG_W / BX, IMG_H);                // 8 x 512 workgroups
    svconv_kernel<<<grid, NTHREADS, 0, stream>>>(x, kern, out);
}